// _global_attention_sop_d_22814866277106
// MI455X (gfx1250) — compile-verified
//
#include <hip/hip_runtime.h>
#include <hip/hip_bf16.h>

#define B_GRAPHS 64
#define N_PER    512
#define D_CH     64
#define D_RED    16            // D / R
#define M_NODES  (B_GRAPHS * N_PER)
#define TRIU     2080          // D*(D+1)/2
#define NPB      4             // nodes per block in triu kernel

typedef float v2f __attribute__((ext_vector_type(2)));
typedef float v8f __attribute__((ext_vector_type(8)));

// ---------------------------------------------------------------------------
// Kernel 1: segment mean.  gap[b][d] = mean_{r<512} x[b*512+r][d]
// 64 blocks x 256 threads; thread (q,d) sums rows q*128..q*128+127 of col d.
// Loads are fully coalesced (each 64-thread group reads one contiguous row).
// ---------------------------------------------------------------------------
__global__ void __launch_bounds__(256)
gap_kernel(const float* __restrict__ x, float* __restrict__ gap) {
  __shared__ float red[256];
  const int b   = blockIdx.x;
  const int tid = threadIdx.x;
  const int d   = tid & 63;
  const int q   = tid >> 6;
  const float* px = x + ((size_t)b * N_PER + (size_t)q * 128) * D_CH + d;
  float s = 0.f;
#pragma unroll 8
  for (int r = 0; r < 128; ++r) s += px[(size_t)r * D_CH];
  red[tid] = s;
  __syncthreads();
  if (tid < 64) {
    gap[b * D_CH + tid] =
        (red[tid] + red[64 + tid] + red[128 + tid] + red[192 + tid]) * (1.0f / 512.0f);
  }
}

// ---------------------------------------------------------------------------
// Kernel 2: imp = sigmoid( leaky_relu(gap @ W1^T + b1) @ W2^T + b2 )
// Single workgroup, 16 wave32.  Both GEMMs run on V_WMMA_F32_16X16X4_F32.
//
// ISA operand layouts (f32, wave32):
//   A 16x4 : lane L -> row M=L%16 ; VGPR v holds K = v + 2*(L>=16)
//   B 4x16 : lane L -> col N=L%16 ; VGPR v holds K = v + 2*(L>=16)
//   C/D    : VGPR v, lane L -> (M = v + 8*(L>=16), N = L%16)
// ---------------------------------------------------------------------------
__global__ void __launch_bounds__(512)
mlp_kernel(const float* __restrict__ gap,
           const float* __restrict__ W1, const float* __restrict__ b1,
           const float* __restrict__ W2, const float* __restrict__ b2,
           float* __restrict__ imp) {
  __shared__ float hlds[64 * D_RED];     // h [64,16]
  const int tid  = threadIdx.x;
  const int wave = tid >> 5;
  const int lane = tid & 31;
  const int hi   = (lane >> 4) & 1;
  const int l16  = lane & 15;

  // ---- GEMM1: gap[64,64] @ W1^T[64,16] -> h[64,16]; waves 0..3, M-tiles ----
  if (wave < 4) {
    const int m0 = wave * 16;
    v8f c = {};
#pragma unroll
    for (int s = 0; s < 16; ++s) {          // K = 64 in steps of 4
      const int kk = 4 * s + 2 * hi;
      v2f a, bm;
      a.x  = gap[(m0 + l16) * D_CH + kk];
      a.y  = gap[(m0 + l16) * D_CH + kk + 1];
      bm.x = W1[l16 * D_CH + kk];           // B[k][j] = W1[j][k]
      bm.y = W1[l16 * D_CH + kk + 1];
      c = __builtin_amdgcn_wmma_f32_16x16x4_f32(false, a, false, bm,
                                                (short)0, c, false, false);
    }
    const float bias = b1[l16];
#pragma unroll
    for (int v = 0; v < 8; ++v) {
      float y = c[v] + bias;
      y = (y > 0.f) ? y : 0.01f * y;        // leaky_relu(0.01)
      hlds[(m0 + v + 8 * hi) * D_RED + l16] = y;
    }
  }
  __syncthreads();

  // ---- GEMM2: h[64,16] @ W2^T[16,64] -> imp[64,64]; 16 waves, 16x16 tiles ----
  {
    const int m0 = (wave >> 2) * 16;
    const int n0 = (wave & 3) * 16;
    v8f c = {};
#pragma unroll
    for (int s = 0; s < 4; ++s) {           // K = 16 in steps of 4
      const int kk = 4 * s + 2 * hi;
      v2f a, bm;
      a.x  = hlds[(m0 + l16) * D_RED + kk];
      a.y  = hlds[(m0 + l16) * D_RED + kk + 1];
      bm.x = W2[(n0 + l16) * D_RED + kk];   // B[k][j] = W2[j][k]
      bm.y = W2[(n0 + l16) * D_RED + kk + 1];
      c = __builtin_amdgcn_wmma_f32_16x16x4_f32(false, a, false, bm,
                                                (short)0, c, false, false);
    }
    const float bias = b2[n0 + l16];
#pragma unroll
    for (int v = 0; v < 8; ++v) {
      float y = c[v] + bias;
      y = 1.0f / (1.0f + __expf(-y));       // sigmoid
      imp[(m0 + v + 8 * hi) * D_CH + n0 + l16] = y;
    }
  }
}

// ---------------------------------------------------------------------------
// Kernel 3: out[k, t] = xg[k,i(t)] * xg[k,j(t)],  xg = x * imp[batch]
// 4 nodes / 256-thread block.  Gated features + packed (i,j) table in LDS,
// float4 stores -> coalesced global_store_b128 (store-BW bound, ~273 MB).
// ---------------------------------------------------------------------------
__global__ void __launch_bounds__(256)
triu_kernel(const float* __restrict__ x, const int* __restrict__ batch,
            const float* __restrict__ imp, float* __restrict__ out) {
  __shared__ float          xg[NPB * D_CH];
  __shared__ unsigned short pr_tab[TRIU];
  const int tid = threadIdx.x;
  const int k0  = blockIdx.x * NPB;

  // build row-major triu (i,j) table: thread i fills row i (i<=j<64)
  if (tid < 64) {
    const int i   = tid;
    const int off = i * D_CH - (i * (i - 1)) / 2;
    for (int j = i; j < D_CH; ++j)
      pr_tab[off + (j - i)] = (unsigned short)((i << 8) | j);
  }

  // gated features for the block's NPB nodes (coalesced 1 KB read)
  {
    const int node = k0 + (tid >> 6);
    const int d    = tid & 63;
    const int b    = batch[node];
    xg[tid] = x[(size_t)k0 * D_CH + tid] * imp[b * D_CH + d];
  }
  __syncthreads();

  // NPB * 520 float4 outputs per block
  for (int u = tid; u < NPB * (TRIU / 4); u += 256) {
    const int node = u / (TRIU / 4);
    const int t4   = u - node * (TRIU / 4);
    const unsigned short p0 = pr_tab[4 * t4 + 0];
    const unsigned short p1 = pr_tab[4 * t4 + 1];
    const unsigned short p2 = pr_tab[4 * t4 + 2];
    const unsigned short p3 = pr_tab[4 * t4 + 3];
    const float* xn = &xg[node * D_CH];
    float4 o;
    o.x = xn[p0 >> 8] * xn[p0 & 255];
    o.y = xn[p1 >> 8] * xn[p1 & 255];
    o.z = xn[p2 >> 8] * xn[p2 & 255];
    o.w = xn[p3 >> 8] * xn[p3 & 255];
    *(float4*)(out + (size_t)(k0 + node) * TRIU + 4 * t4) = o;
  }
}

// ---------------------------------------------------------------------------
extern "C" void kernel_launch(void* const* d_in, const int* in_sizes, int n_in,
                              void* d_out, int out_size, void* d_ws, size_t ws_size,
                              hipStream_t stream) {
  const float* x     = (const float*)d_in[0];
  const int*   batch = (const int*)d_in[1];
  const float* W1    = (const float*)d_in[2];
  const float* b1    = (const float*)d_in[3];
  const float* W2    = (const float*)d_in[4];
  const float* b2    = (const float*)d_in[5];
  float* out = (float*)d_out;

  float* gap = (float*)d_ws;               // [64,64]
  float* imp = gap + B_GRAPHS * D_CH;      // [64,64]

  gap_kernel<<<B_GRAPHS, 256, 0, stream>>>(x, gap);
  mlp_kernel<<<1, 512, 0, stream>>>(gap, W1, b1, W2, b2, imp);
  triu_kernel<<<M_NODES / NPB, 256, 0, stream>>>(x, batch, imp, out);
}